// AdversarialSinkhornDivergence_62620623176161
// MI455X (gfx1250) — compile-verified
//
#include <hip/hip_runtime.h>
#include <math.h>

#define NROWS 8192
#define NC    10
#define KPAD  12
#define NT    512        // NROWS / 16
#define EPSV  0.1f
#define INV_EPS 10.0f
#define THRESHV 0.1f
#define MAX_ITER 50

typedef __attribute__((ext_vector_type(2))) float v2f;
typedef __attribute__((ext_vector_type(8))) float v8f;

// ---- workspace layout (float offsets) ----
#define WS_XPC   0                        // padded clean  [8192][12]
#define WS_XPA   (NROWS*KPAD)             // padded adv    [8192][12]
#define WS_NORMC (2*NROWS*KPAD)           // ||clean_i||^2
#define WS_NORMA (WS_NORMC + NROWS)       // ||adv_i||^2
#define WS_U     (WS_NORMA + NROWS)       // u[3][8192]
#define WS_V     (WS_U + 3*NROWS)         // v[3][8192]
#define WS_ERR   (WS_V + 3*NROWS)         // err[3] (+pad)
#define WS_DIST  (WS_ERR + 4)             // dist[3] (+pad)
#define WS_DONE  (WS_DIST + 4)            // done[3] as int
#define WS_END   (WS_DONE + 4)

__device__ __forceinline__ float logmu() { return __logf(1.0f/(float)NROWS + 1e-8f); }

// instance z: 0 -> (clean, adv); 1 -> (adv, adv); 2 -> (clean, clean)
__device__ __forceinline__ const float* selX (float* ws, int z){ return ws + ((z==1) ? WS_XPA   : WS_XPC  ); }
__device__ __forceinline__ const float* selY (float* ws, int z){ return ws + ((z==2) ? WS_XPC   : WS_XPA  ); }
__device__ __forceinline__ const float* selNX(float* ws, int z){ return ws + ((z==1) ? WS_NORMA : WS_NORMC); }
__device__ __forceinline__ const float* selNY(float* ws, int z){ return ws + ((z==2) ? WS_NORMC : WS_NORMA); }

// ------------------------------------------------------------------ init
__global__ void init_kernel(float* ws, float* out) {
  int n = WS_END - WS_U;
  for (int i = blockIdx.x*blockDim.x + threadIdx.x; i < n; i += gridDim.x*blockDim.x)
    ws[WS_U + i] = 0.0f;                 // zeroes u, v, err, dist, done
  if (blockIdx.x == 0 && threadIdx.x < 2) out[threadIdx.x] = 0.0f;
}

// ------------------------------------------------------- pad + row norms
__global__ void pad_norm_kernel(const float* __restrict__ clean,
                                const float* __restrict__ adv, float* ws) {
  int idx = blockIdx.x*blockDim.x + threadIdx.x;
  if (idx >= 2*NROWS) return;
  bool isC = idx < NROWS;
  int row = isC ? idx : idx - NROWS;
  const float* src = isC ? clean : adv;
  float* dst = ws + (isC ? WS_XPC : WS_XPA) + row*KPAD;
  float n = 0.0f;
#pragma unroll
  for (int k = 0; k < NC; ++k) { float x = src[row*NC + k]; dst[k] = x; n += x*x; }
  dst[10] = 0.0f; dst[11] = 0.0f;
  ws[(isC ? WS_NORMC : WS_NORMA) + row] = n;
}

// --------------------------------------------------------- cross entropy
__global__ void ce_kernel(const float* __restrict__ adv,
                          const int* __restrict__ tgt, float* out) {
  int i = blockIdx.x*blockDim.x + threadIdx.x;
  if (i >= NROWS) return;
  const float* r = adv + i*NC;
  float mx = r[0];
#pragma unroll
  for (int k = 1; k < NC; ++k) mx = fmaxf(mx, r[k]);
  float s = 0.0f;
#pragma unroll
  for (int k = 0; k < NC; ++k) s += __expf(r[k] - mx);
  float logp = r[tgt[i]] - mx - __logf(s);
  atomicAdd(out, -logp * (1.0f/(float)NROWS));
}

// -------------------------------------------------- u update (row LSE)
__global__ __launch_bounds__(256)
void sinkhorn_u_kernel(float* ws) {
  const int z = blockIdx.y;
  int* done = (int*)(ws + WS_DONE);
  if (done[z]) return;
  const int lane = threadIdx.x & 31, wave = threadIdx.x >> 5;
  const int rb = blockIdx.x*8 + wave;        // row block (16 rows)
  const int r16 = lane & 15, hi = lane >> 4;

  const float* Xp = selX(ws, z); const float* Yp = selY(ws, z);
  const float* nx = selNX(ws, z); const float* ny = selNY(ws, z);
  float* u = ws + WS_U + z*NROWS;
  const float* v = ws + WS_V + z*NROWS;

  const float* xr = Xp + (rb*16 + r16)*KPAD + hi*2;   // A frag: rows fixed
  v2f a0 = *(const v2f*)(xr);
  v2f a1 = *(const v2f*)(xr + 4);
  v2f a2 = *(const v2f*)(xr + 8);

  float u_r[8], p_r[8], m[8], s[8];
#pragma unroll
  for (int r = 0; r < 8; ++r) {
    int row = rb*16 + r + hi*8;
    u_r[r] = u[row]; p_r[r] = u_r[r] - nx[row];
    m[r] = -__builtin_inff(); s[r] = 0.0f;
  }

  for (int jt = 0; jt < NT; ++jt) {
    const float* yr = Yp + (jt*16 + r16)*KPAD + hi*2;
    v2f b0 = *(const v2f*)(yr);
    v2f b1 = *(const v2f*)(yr + 4);
    v2f b2 = *(const v2f*)(yr + 8);
    v8f acc = {0.f,0.f,0.f,0.f,0.f,0.f,0.f,0.f};
    acc = __builtin_amdgcn_wmma_f32_16x16x4_f32(false, a0, false, b0, (short)0, acc, false, false);
    acc = __builtin_amdgcn_wmma_f32_16x16x4_f32(false, a1, false, b1, (short)0, acc, false, false);
    acc = __builtin_amdgcn_wmma_f32_16x16x4_f32(false, a2, false, b2, (short)0, acc, false, false);
    int col = jt*16 + r16;
    float q = v[col] - ny[col];
#pragma unroll
    for (int r = 0; r < 8; ++r) {
      float t = (p_r[r] + q + 2.0f*acc[r]) * INV_EPS;   // (u+v-C)/eps
      float mn = fmaxf(m[r], t);
      s[r] = s[r]*__expf(m[r] - mn) + __expf(t - mn);
      m[r] = mn;
    }
  }
  // merge the 16 column-lanes within each half (rows stay per-half)
#pragma unroll
  for (int mask = 1; mask <= 8; mask <<= 1) {
#pragma unroll
    for (int r = 0; r < 8; ++r) {
      float mo = __shfl_xor(m[r], mask, 32);
      float so = __shfl_xor(s[r], mask, 32);
      float mn = fmaxf(m[r], mo);
      s[r] = s[r]*__expf(m[r] - mn) + so*__expf(mo - mn);
      m[r] = mn;
    }
  }
  if (r16 == 0) {
    float ediff = 0.0f;
#pragma unroll
    for (int r = 0; r < 8; ++r) {
      int row = rb*16 + r + hi*8;
      float lse = m[r] + __logf(s[r]);
      float un = EPSV*(logmu() - lse) + u_r[r];
      u[row] = un;
      ediff += fabsf(un - u_r[r]);
    }
    atomicAdd(ws + WS_ERR + z, ediff);
  }
}

// -------------------------------------------------- v update (col LSE)
__global__ __launch_bounds__(256)
void sinkhorn_v_kernel(float* ws) {
  const int z = blockIdx.y;
  int* done = (int*)(ws + WS_DONE);
  if (done[z]) return;
  const int lane = threadIdx.x & 31, wave = threadIdx.x >> 5;
  const int cb = blockIdx.x*8 + wave;        // column block (16 cols)
  const int r16 = lane & 15, hi = lane >> 4;

  const float* Xp = selX(ws, z); const float* Yp = selY(ws, z);
  const float* nx = selNX(ws, z); const float* ny = selNY(ws, z);
  const float* u = ws + WS_U + z*NROWS;
  float* v = ws + WS_V + z*NROWS;

  const float* yr = Yp + (cb*16 + r16)*KPAD + hi*2;   // B frag: cols fixed
  v2f b0 = *(const v2f*)(yr);
  v2f b1 = *(const v2f*)(yr + 4);
  v2f b2 = *(const v2f*)(yr + 8);

  int col = cb*16 + r16;
  float vc = v[col];
  float q = vc - ny[col];
  float m = -__builtin_inff(), s = 0.0f;

  for (int it = 0; it < NT; ++it) {
    const float* xr = Xp + (it*16 + r16)*KPAD + hi*2;
    v2f a0 = *(const v2f*)(xr);
    v2f a1 = *(const v2f*)(xr + 4);
    v2f a2 = *(const v2f*)(xr + 8);
    v8f acc = {0.f,0.f,0.f,0.f,0.f,0.f,0.f,0.f};
    acc = __builtin_amdgcn_wmma_f32_16x16x4_f32(false, a0, false, b0, (short)0, acc, false, false);
    acc = __builtin_amdgcn_wmma_f32_16x16x4_f32(false, a1, false, b1, (short)0, acc, false, false);
    acc = __builtin_amdgcn_wmma_f32_16x16x4_f32(false, a2, false, b2, (short)0, acc, false, false);
#pragma unroll
    for (int r = 0; r < 8; ++r) {
      int row = it*16 + r + hi*8;
      float t = (u[row] - nx[row] + q + 2.0f*acc[r]) * INV_EPS;
      float mn = fmaxf(m, t);
      s = s*__expf(m - mn) + __expf(t - mn);
      m = mn;
    }
  }
  // merge the two row-halves that share this column
  float mo = __shfl_xor(m, 16, 32);
  float so = __shfl_xor(s, 16, 32);
  float mn = fmaxf(m, mo);
  s = s*__expf(m - mn) + so*__expf(mo - mn);
  m = mn;
  if (hi == 0) {
    float lse = m + __logf(s);
    v[col] = EPSV*(logmu() - lse) + vc;
  }
}

// -------------------------------------------------- done-flag update
__global__ void sinkhorn_done_kernel(float* ws) {
  int z = threadIdx.x;
  if (z >= 3) return;
  int* done = (int*)(ws + WS_DONE);
  float* err = ws + WS_ERR;
  if (!done[z] && err[z] < THRESHV) done[z] = 1;   // freeze from next iter (matches scan carry)
  err[z] = 0.0f;
}

// -------------------------------------------------- sum(pi * Cm)
__global__ __launch_bounds__(256)
void transport_cost_kernel(float* ws) {
  const int z = blockIdx.y;
  const int lane = threadIdx.x & 31, wave = threadIdx.x >> 5;
  const int rb = blockIdx.x*8 + wave;
  const int r16 = lane & 15, hi = lane >> 4;

  const float* Xp = selX(ws, z); const float* Yp = selY(ws, z);
  const float* nx = selNX(ws, z); const float* ny = selNY(ws, z);
  const float* u = ws + WS_U + z*NROWS;
  const float* v = ws + WS_V + z*NROWS;

  const float* xr = Xp + (rb*16 + r16)*KPAD + hi*2;
  v2f a0 = *(const v2f*)(xr);
  v2f a1 = *(const v2f*)(xr + 4);
  v2f a2 = *(const v2f*)(xr + 8);

  float u_r[8], nx_r[8];
#pragma unroll
  for (int r = 0; r < 8; ++r) {
    int row = rb*16 + r + hi*8;
    u_r[r] = u[row]; nx_r[r] = nx[row];
  }

  float local = 0.0f;
  for (int jt = 0; jt < NT; ++jt) {
    const float* yrp = Yp + (jt*16 + r16)*KPAD + hi*2;
    v2f b0 = *(const v2f*)(yrp);
    v2f b1 = *(const v2f*)(yrp + 4);
    v2f b2 = *(const v2f*)(yrp + 8);
    v8f acc = {0.f,0.f,0.f,0.f,0.f,0.f,0.f,0.f};
    acc = __builtin_amdgcn_wmma_f32_16x16x4_f32(false, a0, false, b0, (short)0, acc, false, false);
    acc = __builtin_amdgcn_wmma_f32_16x16x4_f32(false, a1, false, b1, (short)0, acc, false, false);
    acc = __builtin_amdgcn_wmma_f32_16x16x4_f32(false, a2, false, b2, (short)0, acc, false, false);
    int col = jt*16 + r16;
    float vc = v[col], nyc = ny[col];
#pragma unroll
    for (int r = 0; r < 8; ++r) {
      float cost = nx_r[r] + nyc - 2.0f*acc[r];
      float t = (u_r[r] + vc - cost) * INV_EPS;
      local += __expf(t) * cost;           // pi * Cm (converged: far below overflow)
    }
  }
#pragma unroll
  for (int mask = 1; mask <= 16; mask <<= 1) local += __shfl_xor(local, mask, 32);
  if (lane == 0) atomicAdd(ws + WS_DIST + z, local);
}

// -------------------------------------------------- combine outputs
__global__ void finalize_kernel(float* ws, float* out) {
  if (threadIdx.x == 0) {
    const float* d = ws + WS_DIST;
    out[1] = d[0] - 0.5f*(d[1] + d[2]);   // loss_sink; out[0] = CE already accumulated
  }
}

extern "C" void kernel_launch(void* const* d_in, const int* in_sizes, int n_in,
                              void* d_out, int out_size, void* d_ws, size_t ws_size,
                              hipStream_t stream) {
  const float* clean = (const float*)d_in[0];
  const float* adv   = (const float*)d_in[1];
  const int*   tgt   = (const int*)d_in[2];
  float* out = (float*)d_out;
  float* ws  = (float*)d_ws;

  init_kernel<<<64, 256, 0, stream>>>(ws, out);
  pad_norm_kernel<<<(2*NROWS + 255)/256, 256, 0, stream>>>(clean, adv, ws);
  ce_kernel<<<NROWS/256, 256, 0, stream>>>(adv, tgt, out);

  dim3 grid(NT/8, 3);   // 8 waves/block, 512 stripes per instance, 3 instances
  for (int it = 0; it < MAX_ITER; ++it) {
    sinkhorn_u_kernel<<<grid, 256, 0, stream>>>(ws);
    sinkhorn_v_kernel<<<grid, 256, 0, stream>>>(ws);
    sinkhorn_done_kernel<<<1, 32, 0, stream>>>(ws);
  }
  transport_cost_kernel<<<grid, 256, 0, stream>>>(ws);
  finalize_kernel<<<1, 32, 0, stream>>>(ws, out);
}